// GraphConvolution_2465311228029
// MI455X (gfx1250) — compile-verified
//
#include <hip/hip_runtime.h>
#include <hip/hip_bf16.h>

typedef __attribute__((ext_vector_type(2))) float v2f;
typedef __attribute__((ext_vector_type(8))) float v8f;

#define DF 64

// ---------------------------------------------------------------------------
// Stage 1: y_{low,high,mlp} = x @ W_{low,high,mlp} using V_WMMA_F32_16X16X4_F32
// grid.x = row tiles of 32, grid.y = 3 (which weight), block = 256 (8 waves).
// Each wave computes one 16x16 f32 tile, chaining 16 WMMAs over K=64.
// ---------------------------------------------------------------------------
__global__ void gemm3_wmma(const float* __restrict__ x,
                           const float* __restrict__ Wl,
                           const float* __restrict__ Wh,
                           const float* __restrict__ Wm,
                           float* __restrict__ yl,
                           float* __restrict__ yh,
                           float* __restrict__ ym,
                           int nrows)
{
    const float* W;
    float* y;
    if (blockIdx.y == 0)      { W = Wl; y = yl; }
    else if (blockIdx.y == 1) { W = Wh; y = yh; }
    else                      { W = Wm; y = ym; }

    const int wave = threadIdx.x >> 5;     // 0..7
    const int lane = threadIdx.x & 31;
    const int l15  = lane & 15;
    const int half = lane >> 4;            // 0: K pair {0,1}, 1: K pair {2,3}

    const int rowbase = blockIdx.x * 32 + (wave >> 2) * 16;
    const int colbase = (wave & 3) * 16;

    int arow = rowbase + l15;              // A: M = lane%16
    if (arow >= nrows) arow = nrows - 1;   // clamp loads; stores guarded below
    const int col = colbase + l15;         // B/D: N = lane%16

    const float* __restrict__ xrow = x + (size_t)arow * DF;

    v8f c = {0.f, 0.f, 0.f, 0.f, 0.f, 0.f, 0.f, 0.f};
#pragma unroll
    for (int kk = 0; kk < 16; ++kk) {
        const int ka = kk * 4 + half * 2;
        v2f a = *(const v2f*)(xrow + ka);                 // A[M, ka..ka+1]
        v2f b;
        b.x = W[(size_t)ka * DF + col];                   // B[ka,   N]
        b.y = W[(size_t)(ka + 1) * DF + col];             // B[ka+1, N]
        c = __builtin_amdgcn_wmma_f32_16x16x4_f32(
                /*neg_a=*/false, a, /*neg_b=*/false, b,
                /*c_mod=*/(short)0, c, /*reuse_a=*/false, /*reuse_b=*/false);
    }

#pragma unroll
    for (int v = 0; v < 8; ++v) {          // D: VGPR v -> M = v + 8*half
        const int orow = rowbase + v + 8 * half;
        if (orow < nrows) y[(size_t)orow * DF + col] = c[v];
    }
}

// ---------------------------------------------------------------------------
// Stage 2: acc_{low,high}[dst] += val * xw_{low,high}[src]  (wave per edge)
// ---------------------------------------------------------------------------
__global__ void spmm_edges(const float* __restrict__ xwl,
                           const float* __restrict__ xwh,
                           const float* __restrict__ vl,
                           const float* __restrict__ vh,
                           const int* __restrict__ esrc,
                           const int* __restrict__ edst,
                           float* __restrict__ accl,
                           float* __restrict__ acch,
                           int E)
{
    const int e    = (int)((blockIdx.x * blockDim.x + threadIdx.x) >> 5);
    const int lane = threadIdx.x & 31;
    if (e >= E) return;

    const int   s = esrc[e];
    const int   d = edst[e];
    const float a = vl[e];
    const float b = vh[e];

    const v2f fl = *(const v2f*)(xwl + (size_t)s * DF + lane * 2);
    const v2f fh = *(const v2f*)(xwh + (size_t)s * DF + lane * 2);

    float* ol = accl + (size_t)d * DF + lane * 2;
    float* oh = acch + (size_t)d * DF + lane * 2;
    unsafeAtomicAdd(ol,     a * fl.x);
    unsafeAtomicAdd(ol + 1, a * fl.y);
    unsafeAtomicAdd(oh,     b * fh.x);
    unsafeAtomicAdd(oh + 1, b * fh.y);
}

// ---------------------------------------------------------------------------
// Stage 3: per-edge fake label + masked segment sums, ReLU fused at gather.
// neigh layout (6 arrays of N*64): [hete_lo, hete_hi, homo_lo, homo_hi,
//                                   unk_lo,  unk_hi]
// ---------------------------------------------------------------------------
__global__ void agg_edges(const float* __restrict__ accl,
                          const float* __restrict__ acch,
                          const int* __restrict__ esrc,
                          const int* __restrict__ edst,
                          const int* __restrict__ lab,
                          float* __restrict__ neigh,
                          int E, int N)
{
    const int e    = (int)((blockIdx.x * blockDim.x + threadIdx.x) >> 5);
    const int lane = threadIdx.x & 31;
    if (e >= E) return;

    const int s  = esrc[e];
    const int d  = edst[e];
    const int sl = lab[s];
    const int dl = lab[d];
    const int fake = ((sl < 0) || (dl < 0)) ? 2 : ((sl != dl) ? 1 : 0);
    const int j    = (fake == 1) ? 0 : ((fake == 0) ? 2 : 4);

    const size_t NF = (size_t)N * DF;
    const v2f hl = *(const v2f*)(accl + (size_t)s * DF + lane * 2);
    const v2f hh = *(const v2f*)(acch + (size_t)s * DF + lane * 2);

    float* o = neigh + (size_t)j * NF + (size_t)d * DF + lane * 2;
    unsafeAtomicAdd(o,          fmaxf(hl.x, 0.f));
    unsafeAtomicAdd(o + 1,      fmaxf(hl.y, 0.f));
    unsafeAtomicAdd(o + NF,     fmaxf(hh.x, 0.f));
    unsafeAtomicAdd(o + NF + 1, fmaxf(hh.y, 0.f));
}

// ---------------------------------------------------------------------------
// Stage 4: attention7 + softmax + weighted combine (wave per node, wave32)
// ---------------------------------------------------------------------------
__global__ void finalize_att(const float* __restrict__ neigh,   // 6 * N*64
                             const float* __restrict__ xwm,     // pre-ReLU mlp
                             const float* __restrict__ avlf,
                             const float* __restrict__ avhf,
                             const float* __restrict__ avlb,
                             const float* __restrict__ avhb,
                             const float* __restrict__ avlu,
                             const float* __restrict__ avhu,
                             const float* __restrict__ avm,
                             const float* __restrict__ att7,    // [7,7]
                             float* __restrict__ out,
                             int N)
{
    const int node = (int)((blockIdx.x * blockDim.x + threadIdx.x) >> 5);
    const int lane = threadIdx.x & 31;
    if (node >= N) return;

    const size_t NF = (size_t)N * DF;
    const int d0 = lane * 2;

    const float* avs[7] = {avlf, avhf, avlb, avhb, avlu, avhu, avm};
    float a0[7], a1[7], p[7];

#pragma unroll
    for (int j = 0; j < 6; ++j) {
        const float* base = neigh + (size_t)j * NF + (size_t)node * DF + d0;
        a0[j] = base[0];
        a1[j] = base[1];
    }
    {
        const float* base = xwm + (size_t)node * DF + d0;
        a0[6] = fmaxf(base[0], 0.f);          // output_mlp = relu(x @ W_mlp)
        a1[6] = fmaxf(base[1], 0.f);
    }

#pragma unroll
    for (int j = 0; j < 7; ++j)
        p[j] = a0[j] * avs[j][d0] + a1[j] * avs[j][d0 + 1];

    // wave32 reduction: every lane ends with full dot products
#pragma unroll
    for (int off = 16; off > 0; off >>= 1) {
#pragma unroll
        for (int j = 0; j < 7; ++j)
            p[j] += __shfl_xor(p[j], off, 32);
    }

    float feats[7];
#pragma unroll
    for (int j = 0; j < 7; ++j)
        feats[j] = 1.f / (1.f + expf(-p[j]));

    float lg[7];
    float m = -3.0e38f;
#pragma unroll
    for (int c = 0; c < 7; ++c) {
        float sacc = 0.f;
#pragma unroll
        for (int j = 0; j < 7; ++j) sacc += feats[j] * att7[j * 7 + c];
        lg[c] = sacc * (1.0f / 7.0f);
        m = fmaxf(m, lg[c]);
    }
    float denom = 0.f;
#pragma unroll
    for (int c = 0; c < 7; ++c) { lg[c] = expf(lg[c] - m); denom += lg[c]; }

    float o0 = 0.f, o1 = 0.f;
#pragma unroll
    for (int c = 0; c < 7; ++c) {
        const float att = lg[c] / denom;
        o0 += att * a0[c];
        o1 += att * a1[c];
    }
    float* ob = out + (size_t)node * DF + d0;
    ob[0] = 7.f * o0;
    ob[1] = 7.f * o1;
}

// ---------------------------------------------------------------------------
extern "C" void kernel_launch(void* const* d_in, const int* in_sizes, int n_in,
                              void* d_out, int out_size, void* d_ws, size_t ws_size,
                              hipStream_t stream)
{
    const float* x     = (const float*)d_in[0];
    const float* vlow  = (const float*)d_in[1];
    const float* vhigh = (const float*)d_in[2];
    const float* Wl    = (const float*)d_in[3];
    const float* Wh    = (const float*)d_in[4];
    const float* Wm    = (const float*)d_in[5];
    const float* avlf  = (const float*)d_in[6];
    const float* avhf  = (const float*)d_in[7];
    const float* avlb  = (const float*)d_in[8];
    const float* avhb  = (const float*)d_in[9];
    const float* avlu  = (const float*)d_in[10];
    const float* avhu  = (const float*)d_in[11];
    const float* avm   = (const float*)d_in[12];
    const float* att7  = (const float*)d_in[13];
    const int*   esrc  = (const int*)d_in[14];
    const int*   edst  = (const int*)d_in[15];
    const int*   lab   = (const int*)d_in[16];

    const int N = in_sizes[0] / DF;
    const int E = in_sizes[1];
    const size_t NF = (size_t)N * DF;

    float* ws    = (float*)d_ws;
    float* xwl   = ws;                 // x @ W_low
    float* xwh   = ws + NF;            // x @ W_high
    float* xwm   = ws + 2 * NF;        // x @ W_mlp   (pre-ReLU)
    float* accl  = ws + 3 * NF;        // spmm_low    (pre-ReLU)
    float* acch  = ws + 4 * NF;        // spmm_high   (pre-ReLU)
    float* neigh = ws + 5 * NF;        // 6 masked aggregate arrays

    // zero the 8 atomic-accumulated buffers (acc_low..neigh[5])
    hipMemsetAsync(accl, 0, 8 * NF * sizeof(float), stream);

    dim3 gg((N + 31) / 32, 3);
    gemm3_wmma<<<gg, 256, 0, stream>>>(x, Wl, Wh, Wm, xwl, xwh, xwm, N);

    const int eblocks = (E + 7) / 8;   // 8 waves (edges) per 256-thread block
    spmm_edges<<<eblocks, 256, 0, stream>>>(xwl, xwh, vlow, vhigh,
                                            esrc, edst, accl, acch, E);
    agg_edges<<<eblocks, 256, 0, stream>>>(accl, acch, esrc, edst, lab,
                                           neigh, E, N);

    const int nblocks = (N + 7) / 8;   // 8 waves (nodes) per block
    finalize_att<<<nblocks, 256, 0, stream>>>(neigh, xwm,
                                              avlf, avhf, avlb, avhb,
                                              avlu, avhu, avm, att7,
                                              (float*)d_out, N);
}